// RecurrentLTI_49838800503397
// MI455X (gfx1250) — compile-verified
//
#include <hip/hip_runtime.h>

// Problem constants (from reference config)
#define L_SEQ   65536
#define T_CHUNK 256     // time steps per chunk
#define J_CHUNK 256     // number of chunks
#define NMODE   256     // 2x2 blocks -> complex modes
#define NSTATE  512

typedef float v2f __attribute__((ext_vector_type(2)));
typedef float v8f __attribute__((ext_vector_type(8)));

// ---------------------------------------------------------------------------
// 1) Extract per-mode complex params from A_dsc/b_dsc/c and build power table
//    lambda_i = p - i q  (block [[p,q],[-q,p]]), beta_i = b0 + i b1,
//    gamma_i = c0 - i c1.  powtab[m][i] = lambda_i^m for m = 0..T (inclusive).
// ---------------------------------------------------------------------------
__global__ void prep_modes_kernel(const float* __restrict__ A,
                                  const float* __restrict__ b,
                                  const float* __restrict__ c,
                                  float* __restrict__ powRe, float* __restrict__ powIm,
                                  float* __restrict__ bRe,  float* __restrict__ bIm,
                                  float* __restrict__ gRe,  float* __restrict__ gIm,
                                  float* __restrict__ gbRe, float* __restrict__ gbIm,
                                  float* __restrict__ muRe, float* __restrict__ muIm) {
    int i = threadIdx.x;                 // 256 threads, one per mode
    float p = A[(size_t)(2 * i) * NSTATE + 2 * i];
    float q = A[(size_t)(2 * i) * NSTATE + 2 * i + 1];
    float lr = p, li = -q;
    float br = b[2 * i], bi = b[2 * i + 1];
    float gr = c[2 * i], gi = -c[2 * i + 1];
    bRe[i] = br;  bIm[i] = bi;
    gRe[i] = gr;  gIm[i] = gi;
    gbRe[i] = gr * br - gi * bi;         // gamma*beta (for impulse response)
    gbIm[i] = gr * bi + gi * br;
    float pr = 1.0f, pi = 0.0f;
    for (int m = 0; m <= T_CHUNK; ++m) {
        powRe[m * NMODE + i] = pr;
        powIm[m * NMODE + i] = pi;
        float nr = pr * lr - pi * li;
        float ni = pr * li + pi * lr;
        pr = nr; pi = ni;
    }
    muRe[i] = powRe[T_CHUNK * NMODE + i];   // mu = lambda^T (chunk multiplier)
    muIm[i] = powIm[T_CHUNK * NMODE + i];
}

// h[m] = sum_i Re(gamma_i * beta_i * lambda_i^m)
__global__ void build_h_kernel(const float* __restrict__ powRe, const float* __restrict__ powIm,
                               const float* __restrict__ gbRe, const float* __restrict__ gbIm,
                               float* __restrict__ h) {
    int m = threadIdx.x;                 // 256 threads
    const float* pr = &powRe[m * NMODE];
    const float* pm = &powIm[m * NMODE];
    float acc = 0.0f;
    for (int i = 0; i < NMODE; ++i) acc += gbRe[i] * pr[i] - gbIm[i] * pm[i];
    h[m] = acc;
}

// H[t][s] = h[t-s] for t>=s else 0   (T x T lower-triangular Toeplitz)
__global__ void build_H_kernel(const float* __restrict__ h, float* __restrict__ H) {
    int idx = blockIdx.x * blockDim.x + threadIdx.x;   // 65536
    int t = idx >> 8, s = idx & 255;
    H[idx] = (t >= s) ? h[t - s] : 0.0f;
}

// W_B (NSTATE x T): row 2i = Re(lambda^{T-1-t} beta), row 2i+1 = Im(...)
__global__ void build_WB_kernel(const float* __restrict__ powRe, const float* __restrict__ powIm,
                                const float* __restrict__ bRe, const float* __restrict__ bIm,
                                float* __restrict__ WB) {
    int idx = blockIdx.x * blockDim.x + threadIdx.x;   // 512*256
    int r = idx >> 8, t = idx & 255;
    int i = r >> 1;
    int m = T_CHUNK - 1 - t;
    float pr = powRe[m * NMODE + i], pm = powIm[m * NMODE + i];
    float v = ((r & 1) == 0) ? (pr * bRe[i] - pm * bIm[i])
                             : (pr * bIm[i] + pm * bRe[i]);
    WB[idx] = v;
}

// W_C (T x NSTATE): col 2i = Re(gamma*lambda^{t+1}), col 2i+1 = -Im(gamma*lambda^{t+1})
__global__ void build_WC_kernel(const float* __restrict__ powRe, const float* __restrict__ powIm,
                                const float* __restrict__ gRe, const float* __restrict__ gIm,
                                float* __restrict__ WC) {
    int idx = blockIdx.x * blockDim.x + threadIdx.x;   // 256*512
    int t = idx >> 9, cidx = idx & 511;
    int i = cidx >> 1;
    float pr = powRe[(t + 1) * NMODE + i], pm = powIm[(t + 1) * NMODE + i];
    float v = ((cidx & 1) == 0) ? (gRe[i] * pr - gIm[i] * pm)
                                : -(gRe[i] * pm + gIm[i] * pr);
    WC[idx] = v;
}

// U (T x J) row-major: U[t][j] = u[j*T + t]
__global__ void build_U_kernel(const float* __restrict__ u, float* __restrict__ U) {
    int idx = blockIdx.x * blockDim.x + threadIdx.x;   // 65536
    int t = idx >> 8, j = idx & 255;
    U[idx] = u[(j << 8) + t];
}

// ---------------------------------------------------------------------------
// fp32 WMMA GEMM, double-buffered async global->LDS staging:
//   C(MxN) (+)= A(MxK) @ B(KxN), row-major. Block = 256 threads = 8 waves in a
//   4x2 arrangement -> 64x32 block tile. K staged 16 at a time; next panel's
//   GLOBAL_LOAD_ASYNC_TO_LDS_* runs under the current panel's WMMAs
//   (s_wait_asynccnt 0x2 exploits in-order async completion). LDS rows padded
//   (A: 20 floats, B: 36 floats) for conflict-free ds reads while keeping the
//   16B/8B alignment the async b128/b64 writes need.
//   Fragment layouts per ISA 7.12.2 (wave32).
// ---------------------------------------------------------------------------
__global__ void __launch_bounds__(256)
gemm_wmma_f32(const float* __restrict__ A, const float* __restrict__ B,
              float* __restrict__ C, int M, int N, int K, int accumulate) {
    __shared__ __align__(16) float As[2][64][20];   // A panels: 64 rows x 16 k (+4 pad)
    __shared__ __align__(16) float Bs[2][16][36];   // B panels: 16 k x 32 cols (+4 pad)

    int tid    = threadIdx.x;        // 256 threads
    int waveId = tid >> 5;           // 0..7
    int lane   = tid & 31;
    int tilesN = N >> 5;             // # of 32-wide block tiles
    int brow0  = (blockIdx.x / tilesN) << 6;   // 64-row block tiles
    int bcol0  = (blockIdx.x % tilesN) << 5;
    int wr     = (waveId >> 1) << 4; // wave row offset: 0|16|32|48
    int wc     = (waveId & 1) << 4;  // wave col offset: 0|16

    int half = lane >> 4;            // 0: lanes 0-15, 1: lanes 16-31
    int l15  = lane & 15;
    int kA   = half * 2;             // A/B frag: VGPR0={K0|K2}, VGPR1={K1|K3}
    int mC   = half * 8;             // C/D frag: VGPR v -> M=v+8*half

    // async-copy thread mapping (uniform: every thread issues A b128 + B b64)
    int arow = tid >> 2, acol = (tid & 3) << 2;   // A panel 64x16, 4 floats/thread
    int brow = tid >> 4, bcol = (tid & 15) << 1;  // B panel 16x32, 2 floats/thread
    // low 32 bits of a generic LDS pointer == LDS byte offset (ISA 10.2)
    unsigned ldsA[2] = { (unsigned)(size_t)&As[0][arow][acol],
                         (unsigned)(size_t)&As[1][arow][acol] };
    unsigned ldsB[2] = { (unsigned)(size_t)&Bs[0][brow][bcol],
                         (unsigned)(size_t)&Bs[1][brow][bcol] };

    const float* gaBase = &A[(size_t)(brow0 + arow) * K + acol];
    const float* gbBase = &B[(size_t)brow * N + bcol0 + bcol];
    size_t gbStride = (size_t)16 * N;   // 16 k-rows per stage

    v8f acc = {};
    if (accumulate) {
#pragma unroll
        for (int v = 0; v < 8; ++v)
            acc[v] = C[(size_t)(brow0 + wr + mC + v) * N + bcol0 + wc + l15];
    }

    // prologue: stage 0 -> buffer 0
    asm volatile("global_load_async_to_lds_b128 %0, %1, off"
                 :: "v"(ldsA[0]), "v"((unsigned long long)(size_t)gaBase) : "memory");
    asm volatile("global_load_async_to_lds_b64 %0, %1, off"
                 :: "v"(ldsB[0]), "v"((unsigned long long)(size_t)gbBase) : "memory");

    int nStages = K >> 4;
    for (int s = 0; s < nStages; ++s) {
        int cur = s & 1;
        if (s + 1 < nStages) {
            // issue next panel into the other buffer (overlaps with WMMAs below)
            const float* ga = gaBase + (s + 1) * 16;
            const float* gb = gbBase + (s + 1) * gbStride;
            asm volatile("global_load_async_to_lds_b128 %0, %1, off"
                         :: "v"(ldsA[cur ^ 1]), "v"((unsigned long long)(size_t)ga) : "memory");
            asm volatile("global_load_async_to_lds_b64 %0, %1, off"
                         :: "v"(ldsB[cur ^ 1]), "v"((unsigned long long)(size_t)gb) : "memory");
            // async loads complete in order: <=2 outstanding => stage s landed
            asm volatile("s_wait_asynccnt 0x2" ::: "memory");
        } else {
            asm volatile("s_wait_asynccnt 0x0" ::: "memory");
        }
        __syncthreads();             // stage s visible to all waves
#pragma unroll
        for (int kk = 0; kk < 16; kk += 4) {
            v2f af;
            af.x = As[cur][wr + l15][kk + kA];
            af.y = As[cur][wr + l15][kk + kA + 1];
            v2f bf;
            bf.x = Bs[cur][kk + kA][wc + l15];
            bf.y = Bs[cur][kk + kA + 1][wc + l15];
            acc = __builtin_amdgcn_wmma_f32_16x16x4_f32(
                /*neg_a=*/false, af, /*neg_b=*/false, bf,
                /*c_mod=*/(short)0, acc, /*reuse_a=*/false, /*reuse_b=*/false);
        }
        __syncthreads();             // done reading buf 'cur' before stage s+2 lands
    }

#pragma unroll
    for (int v = 0; v < 8; ++v)
        C[(size_t)(brow0 + wr + mC + v) * N + bcol0 + wc + l15] = acc[v];
}

// ---------------------------------------------------------------------------
// Inter-chunk scan: per mode i, s_in[0]=0; s_in[j] = mu*s_in[j-1] + s_end[j-1]
// 256 modes in parallel, only 256 serial steps (vs 65536 in the reference).
// ---------------------------------------------------------------------------
__global__ void chunk_scan_kernel(const float* __restrict__ Send,
                                  const float* __restrict__ muRe, const float* __restrict__ muIm,
                                  float* __restrict__ Sin) {
    int i = threadIdx.x;                 // 256 threads, one per mode
    float mr = muRe[i], mi = muIm[i];
    const float* er = &Send[(size_t)(2 * i) * J_CHUNK];
    const float* ei = &Send[(size_t)(2 * i + 1) * J_CHUNK];
    float* orr = &Sin[(size_t)(2 * i) * J_CHUNK];
    float* oii = &Sin[(size_t)(2 * i + 1) * J_CHUNK];
    float sr = 0.0f, si = 0.0f;
    for (int j = 0; j < J_CHUNK; ++j) {
        orr[j] = sr; oii[j] = si;
        float nr = mr * sr - mi * si + er[j];
        float ni = mr * si + mi * sr + ei[j];
        sr = nr; si = ni;
    }
}

// y[k] = Y[t][j] + d*u[k],  k = j*T + t
__global__ void finalize_kernel(const float* __restrict__ Y, const float* __restrict__ u,
                                const float* __restrict__ dscal, float* __restrict__ out) {
    int k = blockIdx.x * blockDim.x + threadIdx.x;   // 65536
    int j = k >> 8, t = k & 255;
    out[k] = Y[t * J_CHUNK + j] + dscal[0] * u[k];
}

extern "C" void kernel_launch(void* const* d_in, const int* in_sizes, int n_in,
                              void* d_out, int out_size, void* d_ws, size_t ws_size,
                              hipStream_t stream) {
    (void)in_sizes; (void)n_in; (void)out_size; (void)ws_size;
    const float* u  = (const float*)d_in[0];
    const float* Ad = (const float*)d_in[1];
    const float* bd = (const float*)d_in[2];
    const float* cv = (const float*)d_in[3];
    const float* dv = (const float*)d_in[4];
    float* out = (float*)d_out;

    // Carve workspace (floats), 256B-aligned chunks
    float* w = (float*)d_ws;
    size_t off = 0;
    auto carve = [&](size_t n) { float* p = w + off; off += (n + 63) & ~(size_t)63; return p; };
    float* powRe = carve((T_CHUNK + 1) * NMODE);
    float* powIm = carve((T_CHUNK + 1) * NMODE);
    float* bRe = carve(NMODE); float* bIm = carve(NMODE);
    float* gRe = carve(NMODE); float* gIm = carve(NMODE);
    float* gbRe = carve(NMODE); float* gbIm = carve(NMODE);
    float* muRe = carve(NMODE); float* muIm = carve(NMODE);
    float* h    = carve(T_CHUNK);
    float* Hm   = carve(T_CHUNK * T_CHUNK);        // 256x256
    float* WB   = carve(NSTATE * T_CHUNK);         // 512x256
    float* WC   = carve(T_CHUNK * NSTATE);         // 256x512
    float* Um   = carve(T_CHUNK * J_CHUNK);        // 256x256
    float* Send = carve(NSTATE * J_CHUNK);         // 512x256
    float* Sin  = carve(NSTATE * J_CHUNK);         // 512x256
    float* Ym   = carve(T_CHUNK * J_CHUNK);        // 256x256

    // Mode extraction + power table
    prep_modes_kernel<<<1, NMODE, 0, stream>>>(Ad, bd, cv, powRe, powIm,
                                               bRe, bIm, gRe, gIm, gbRe, gbIm, muRe, muIm);
    build_h_kernel<<<1, T_CHUNK, 0, stream>>>(powRe, powIm, gbRe, gbIm, h);
    build_H_kernel<<<(T_CHUNK * T_CHUNK) / 256, 256, 0, stream>>>(h, Hm);
    build_WB_kernel<<<(NSTATE * T_CHUNK) / 256, 256, 0, stream>>>(powRe, powIm, bRe, bIm, WB);
    build_WC_kernel<<<(T_CHUNK * NSTATE) / 256, 256, 0, stream>>>(powRe, powIm, gRe, gIm, WC);
    build_U_kernel<<<L_SEQ / 256, 256, 0, stream>>>(u, Um);

    // S_end(512x256) = W_B(512x256) @ U(256x256)
    {
        int blocks = (NSTATE / 64) * (J_CHUNK / 32);   // 64 blocks of 64x32 tiles
        gemm_wmma_f32<<<blocks, 256, 0, stream>>>(WB, Um, Send, NSTATE, J_CHUNK, T_CHUNK, 0);
    }
    // Inter-chunk scan (256 serial steps, 256-way mode-parallel)
    chunk_scan_kernel<<<1, NMODE, 0, stream>>>(Send, muRe, muIm, Sin);
    // Y(256x256) = H(256x256) @ U(256x256)
    {
        int blocks = (T_CHUNK / 64) * (J_CHUNK / 32);  // 32 blocks
        gemm_wmma_f32<<<blocks, 256, 0, stream>>>(Hm, Um, Ym, T_CHUNK, J_CHUNK, T_CHUNK, 0);
    }
    // Y += W_C(256x512) @ S_in(512x256)
    {
        int blocks = (T_CHUNK / 64) * (J_CHUNK / 32);  // 32 blocks
        gemm_wmma_f32<<<blocks, 256, 0, stream>>>(WC, Sin, Ym, T_CHUNK, J_CHUNK, NSTATE, 1);
    }
    // y[k] = Y[t][j] + d*u[k]
    finalize_kernel<<<L_SEQ / 256, 256, 0, stream>>>(Ym, u, dv, out);
}